// VariationalNetwork_80418967650926
// MI455X (gfx1250) — compile-verified
//
#include <hip/hip_runtime.h>
#include <hip/hip_bf16.h>

// ---------------------------------------------------------------------------
// Variational network: x = x0 - A*data_term - sum_k c_k * reg_k
//   c_k = w_k*alpha_k,  w_9=1, w_j = 1 + mu[j+1]*w_{j+1},  A = sum c_k
//   data_term = Re( ifft2( mask*fft2(x0) - s ) )          (ortho norm)
//   reg_k     = conv_flip( phi_k * conv(x0, W_k) )        (SAME, zero pad)
//   phi uses mag = clip(sqrt(f^T G f / T + 1e-8),0,1.5) via Gram matrix G.
// WMMA: conv1 as implicit GEMM with V_WMMA_F32_16X16X4_F32
//   M=16 (2 layers x 8 filters), K=9->12 taps (3 chunks), N=16 pixels.
// TDM: interior x0 tiles staged to LDS via tensor_load_to_lds (D# per ISA §8).
// ---------------------------------------------------------------------------

typedef __attribute__((ext_vector_type(2))) float v2f;
typedef __attribute__((ext_vector_type(8))) float v8f;
typedef __attribute__((ext_vector_type(4))) unsigned int u32x4;
typedef __attribute__((ext_vector_type(4))) int i32x4;
typedef __attribute__((ext_vector_type(8))) int i32x8;

#define TT 64
#define HH 512
#define WW 512
#define NPIX (HH * WW)
#define ORTHO 0.04419417382415922f /* 1/sqrt(512) */

// ---------------------------- coefficients ---------------------------------
__global__ void vn_weights_kernel(const float* __restrict__ alpha,
                                  const float* __restrict__ mu,
                                  float* __restrict__ coef) {
  if (threadIdx.x == 0 && blockIdx.x == 0) {
    float w[10];
    w[9] = 1.0f;
    for (int j = 8; j >= 0; --j) w[j] = 1.0f + mu[j + 1] * w[j + 1];
    float A = 0.0f;
    for (int j = 0; j < 10; ++j) {
      float c = w[j] * alpha[j];
      coef[j] = c;
      A += c;
    }
    coef[10] = A;
  }
}

// ---------------------------- 512-pt Stockham FFT ---------------------------
__device__ __forceinline__ float2* fft512(float2* s0, float2* s1, int tid, float dir) {
  float2* src = s0;
  float2* dst = s1;
  int p = 1;
  for (int st = 0; st < 9; ++st) {
    int k = tid & (p - 1);
    float2 u0 = src[tid];
    float2 u1 = src[tid + 256];
    float ang = dir * 3.14159265358979323846f * (float)k / (float)p;
    float sn, cs;
    __sincosf(ang, &sn, &cs);
    float2 tw;
    tw.x = u1.x * cs - u1.y * sn;
    tw.y = u1.x * sn + u1.y * cs;
    int j = ((tid - k) << 1) + k;
    dst[j] = make_float2(u0.x + tw.x, u0.y + tw.y);
    dst[j + p] = make_float2(u0.x - tw.x, u0.y - tw.y);
    p <<= 1;
    float2* t = src; src = dst; dst = t;
    __syncthreads();
  }
  return src;  // result buffer
}

// FFT along W (rows), real input -> complex, ortho-scaled by 1/sqrt(512)
__global__ __launch_bounds__(256) void vn_fft_rows_fwd(const float* __restrict__ x0,
                                                       float2* __restrict__ cplx) {
  __shared__ float2 fa[512];
  __shared__ float2 fb[512];
  const int tid = threadIdx.x;
  const size_t row = blockIdx.x;  // t*512 + h
  fa[tid] = make_float2(x0[row * WW + tid], 0.0f);
  fa[tid + 256] = make_float2(x0[row * WW + tid + 256], 0.0f);
  __syncthreads();
  float2* res = fft512(fa, fb, tid, -1.0f);
  cplx[row * WW + tid] = make_float2(res[tid].x * ORTHO, res[tid].y * ORTHO);
  cplx[row * WW + tid + 256] =
      make_float2(res[tid + 256].x * ORTHO, res[tid + 256].y * ORTHO);
}

// FFT along H (cols) + mask*X - s + inverse FFT along H (all ortho-scaled)
__global__ __launch_bounds__(256) void vn_fft_cols_mask(float2* __restrict__ cplx,
                                                        const float* __restrict__ mask,
                                                        const float* __restrict__ sC) {
  __shared__ float2 fa[512];
  __shared__ float2 fb[512];
  const int tid = threadIdx.x;
  const int t = blockIdx.x >> 9;
  const int w = blockIdx.x & 511;
  fa[tid] = cplx[((size_t)t * HH + tid) * WW + w];
  fa[tid + 256] = cplx[((size_t)t * HH + tid + 256) * WW + w];
  __syncthreads();
  float2* res = fft512(fa, fb, tid, -1.0f);
  float2* other = (res == fa) ? fb : fa;
#pragma unroll
  for (int q = 0; q < 2; ++q) {
    int i = tid + q * 256;
    float2 X = res[i];
    X.x *= ORTHO; X.y *= ORTHO;                 // now full ortho fft2 value
    size_t kidx = ((size_t)t * HH + i) * WW + w;
    float mk = mask[kidx];
    float2 r;
    r.x = mk * X.x - sC[kidx * 2 + 0];
    r.y = mk * X.y - sC[kidx * 2 + 1];
    res[i] = r;
  }
  __syncthreads();
  float2* res2 = fft512(res, other, tid, +1.0f);
#pragma unroll
  for (int q = 0; q < 2; ++q) {
    int i = tid + q * 256;
    float2 v = res2[i];
    v.x *= ORTHO; v.y *= ORTHO;
    cplx[((size_t)t * HH + i) * WW + w] = v;
  }
}

// inverse FFT along W (rows), take real part, scale by coef[10] -> acc
__global__ __launch_bounds__(256) void vn_ifft_rows_real(const float2* __restrict__ cplx,
                                                         const float* __restrict__ coef,
                                                         float* __restrict__ acc) {
  __shared__ float2 fa[512];
  __shared__ float2 fb[512];
  const int tid = threadIdx.x;
  const size_t row = blockIdx.x;  // t*512 + h
  fa[tid] = cplx[row * WW + tid];
  fa[tid + 256] = cplx[row * WW + tid + 256];
  __syncthreads();
  float2* res = fft512(fa, fb, tid, +1.0f);
  const float A = coef[10];
  acc[row * WW + tid] = A * res[tid].x * ORTHO;
  acc[row * WW + tid + 256] = A * res[tid + 256].x * ORTHO;
}

// ---------------------------- Gram + activation (phi) -----------------------
__global__ __launch_bounds__(256) void vn_phi_kernel(const float* __restrict__ x0,
                                                     const float* __restrict__ filters,
                                                     const float* __restrict__ actv,
                                                     float* __restrict__ phiB) {
  __shared__ float xs[18][18];
  const int tid = threadIdx.x;
  const int tx = tid & 15;
  const int ty = tid >> 4;
  const int gx0 = blockIdx.x * 16;
  const int gy0 = blockIdx.y * 16;

  float g[45];
#pragma unroll
  for (int i = 0; i < 45; ++i) g[i] = 0.0f;

  for (int t = 0; t < TT; ++t) {
    __syncthreads();
    for (int i = tid; i < 18 * 18; i += 256) {
      int r = i / 18, c = i % 18;
      int gy = gy0 - 1 + r, gx = gx0 - 1 + c;
      float v = 0.0f;
      if (gy >= 0 && gy < HH && gx >= 0 && gx < WW)
        v = x0[((size_t)t * HH + gy) * WW + gx];
      xs[r][c] = v;
    }
    __syncthreads();
    float p[9];
#pragma unroll
    for (int dy = 0; dy < 3; ++dy)
#pragma unroll
      for (int dx = 0; dx < 3; ++dx) p[dy * 3 + dx] = xs[ty + dy][tx + dx];
    int c45 = 0;
#pragma unroll
    for (int i = 0; i < 9; ++i)
#pragma unroll
      for (int j = i; j < 9; ++j) g[c45++] += p[i] * p[j];
  }

  const int h = gy0 + ty;
  const int w = gx0 + tx;
  for (int k = 0; k < 10; ++k) {
    for (int n = 0; n < 8; ++n) {
      const float* f = filters + (size_t)(k * 8 + n) * 9;
      float fv[9];
#pragma unroll
      for (int i = 0; i < 9; ++i) fv[i] = f[i];
      float v = 0.0f;
      int c45 = 0;
#pragma unroll
      for (int i = 0; i < 9; ++i)
#pragma unroll
        for (int j = i; j < 9; ++j) {
          float term = fv[i] * fv[j] * g[c45++];
          v += (i == j) ? term : 2.0f * term;
        }
      float mag = sqrtf(v * (1.0f / (float)TT) + 1e-8f);
      mag = fminf(mag, 1.5f);
      float pos = mag * (99.0f / 1.5f);
      int i0 = (int)pos;
      i0 = min(i0, 98);
      float fr = pos - (float)i0;
      const float* av = actv + (size_t)(k * 8 + n) * 100;
      float ph = av[i0] + fr * (av[i0 + 1] - av[i0]);
      phiB[((size_t)(k * 8 + n) * HH + h) * WW + w] = ph;
    }
  }
}

// ---------------------------- fused conv pass (WMMA + TDM) ------------------
#define OH 14
#define OW 30
__global__ __launch_bounds__(256) void vn_conv_pair(const float* __restrict__ x0,
                                                    const float* __restrict__ filters,
                                                    const float* __restrict__ phiB,
                                                    const float* __restrict__ coef,
                                                    float* __restrict__ acc,
                                                    int lp) {
  __shared__ float xs[18][34];             // x0 tile + 2 halo, contiguous (TDM tile)
  __shared__ float phis[2][8][16][32];     // phi tiles for both layers
  __shared__ float tmpl[8][16][32];        // tmp tile for one layer
  __shared__ float filtL[2][8][9];         // filters for both layers

  const int k0 = lp * 2;
  const int tid = threadIdx.x;
  const int lane = tid & 31;
  const int wv = tid >> 5;
  const int tileX0 = blockIdx.x * OW;
  const int tileY0 = blockIdx.y * OH;

  if (tid < 144) {
    int kk = tid / 72, rest = tid % 72;
    filtL[kk][rest / 9][rest % 9] = filters[(size_t)(k0 + kk) * 72 + rest];
  }
  const float c0 = coef[k0];
  const float c1 = coef[k0 + 1];

  // phi tiles (clamped coords; invalid positions get zeroed in tmp anyway)
  for (int i = tid; i < 2 * 8 * 16 * 32; i += 256) {
    int col = i & 31;
    int row = (i >> 5) & 15;
    int fn = (i >> 9) & 7;
    int ll = i >> 12;
    int gy = tileY0 - 1 + row, gx = tileX0 - 1 + col;
    int cy = min(max(gy, 0), HH - 1), cx = min(max(gx, 0), WW - 1);
    phis[ll][fn][row][col] = phiB[((size_t)((k0 + ll) * 8 + fn) * HH + cy) * WW + cx];
  }

  // A matrix chunks (16x4 f32 layout: lane m=lane&15, VGPR0 K=2*hi, VGPR1 K=2*hi+1)
  const int mrow = lane & 15;
  const int hi = lane >> 4;
  const int klay = k0 + (mrow >> 3);
  const int filt = mrow & 7;
  v2f A_[3];
#pragma unroll
  for (int c = 0; c < 3; ++c) {
    int t0 = 4 * c + 2 * hi, t1 = t0 + 1;
    A_[c].x = (t0 < 9) ? filters[(size_t)(klay * 8 + filt) * 9 + t0] : 0.0f;
    A_[c].y = (t1 < 9) ? filters[(size_t)(klay * 8 + filt) * 9 + t1] : 0.0f;
  }
  const int nn = lane & 15;

  // Interior blocks can use the Tensor Data Mover (no negative/OOB origin).
  const bool interior = (tileY0 >= 2) && (tileY0 + 15 <= HH - 1) &&
                        (tileX0 >= 2) && (tileX0 + 31 <= WW - 1);
  // D# group1: mask=0, data_size=4B(code 2), no flags; dims per ISA bit layout.
  i32x8 dg1;
  dg1[0] = (2 << 16);          // data_size = 4 bytes
  dg1[1] = (34 << 16);         // tensor_dim0[15:0] (bits 63:48)
  dg1[2] = (18 << 16);         // tensor_dim1[15:0] (bits 95:80)
  dg1[3] = (34 << 16);         // tile_dim0 (bits 127:112)
  dg1[4] = 18;                 // tile_dim1 | tile_dim2=0
  dg1[5] = WW;                 // tensor_dim0_stride lo32
  dg1[6] = 0;
  dg1[7] = 0;
  const i32x4 dzero4 = {0, 0, 0, 0};
  const i32x8 dzero8 = {0, 0, 0, 0, 0, 0, 0, 0};
  const unsigned int ldsAddr = (unsigned int)(unsigned long long)(void*)&xs[0][0];
  const unsigned long long gBase =
      (unsigned long long)(const void*)x0 +
      4ull * ((unsigned long long)(tileY0 - 2) * WW + (unsigned)(tileX0 - 2));
  __syncthreads();

  for (int t = 0; t < TT; ++t) {
    __syncthreads();  // previous tmp/xs consumers done
    // stage x0 tile (rows tileY0-2..+15, cols tileX0-2..+31)
    if (interior) {
      if (wv == 0) {  // one TDM issue per block; TDM ignores EXEC
        unsigned long long ga = gBase + 4ull * (unsigned long long)t * NPIX;
        u32x4 dg0;
        dg0[0] = 1u;                                   // count=1 (valid D#)
        dg0[1] = ldsAddr;                              // lds_addr
        dg0[2] = (unsigned int)(ga & 0xFFFFFFFFull);   // global_addr lo
        dg0[3] = (unsigned int)((ga >> 32) & 0x01FFFFFFull) | (2u << 30);  // hi|type=2
        __builtin_amdgcn_tensor_load_to_lds(dg0, dg1, dzero4, dzero4, dzero8, 0);
        __builtin_amdgcn_s_wait_tensorcnt(0);
      }
    } else {
      for (int i = tid; i < 18 * 34; i += 256) {
        int r = i / 34, c = i % 34;
        int gy = tileY0 - 2 + r, gx = tileX0 - 2 + c;
        float v = 0.0f;
        if (gy >= 0 && gy < HH && gx >= 0 && gx < WW)
          v = x0[((size_t)t * HH + gy) * WW + gx];
        xs[r][c] = v;
      }
    }
    if (t + 1 < TT) {  // hint next slice into cache (global_prefetch_b8)
      int gy = min(max(tileY0, 0), HH - 1);
      __builtin_prefetch(&x0[((size_t)(t + 1) * HH + gy) * WW + min(tileX0, WW - 1)], 0, 1);
    }
    __syncthreads();

    // conv1 as implicit GEMM: 4 WMMA tiles per wave, 3 K-chunks each.
    // Taps >= 9 have A == 0, so B may hold any finite value: clamp tap to 8.
    v8f C_[4];
#pragma unroll
    for (int i = 0; i < 4; ++i) {
      const int tIdx = wv * 4 + i;
      const int ht = tIdx >> 1;
      const int wg = tIdx & 1;
      const int col = wg * 16 + nn;
      v8f cacc = {0.f, 0.f, 0.f, 0.f, 0.f, 0.f, 0.f, 0.f};
#pragma unroll
      for (int c = 0; c < 3; ++c) {
        int t0 = 4 * c + 2 * hi, t1 = t0 + 1;
        int p0 = (t0 < 9) ? t0 : 8;
        int p1 = (t1 < 9) ? t1 : 8;
        v2f B_;
        B_.x = xs[ht + p0 / 3][col + p0 % 3];
        B_.y = xs[ht + p1 / 3][col + p1 % 3];
        cacc = __builtin_amdgcn_wmma_f32_16x16x4_f32(false, A_[c], false, B_,
                                                     (short)0, cacc, false, false);
      }
      C_[i] = cacc;
    }

    float racc0 = 0.0f, racc1 = 0.0f;
#pragma unroll
    for (int ph = 0; ph < 2; ++ph) {
      // scatter this layer's f*phi into tmp (zero outside the image => SAME pad)
#pragma unroll
      for (int i = 0; i < 4; ++i) {
        const int tIdx = wv * 4 + i;
        const int ht = tIdx >> 1;
        const int wg = tIdx & 1;
        const int col = wg * 16 + nn;
        if (hi == ph) {
          int gy = tileY0 - 1 + ht, gx = tileX0 - 1 + col;
          bool valid = (gy >= 0) && (gy < HH) && (gx >= 0) && (gx < WW);
#pragma unroll
          for (int r = 0; r < 8; ++r)
            tmpl[r][ht][col] = valid ? C_[i][r] * phis[ph][r][ht][col] : 0.0f;
        }
      }
      __syncthreads();
      // conv2 (flipped filters == true convolution with W)
      for (int pp = tid; pp < OH * OW; pp += 256) {
        int oh = pp / OW, ow = pp % OW;
        int gy = tileY0 + oh, gx = tileX0 + ow;
        if (gy < HH && gx < WW) {
          float s = 0.0f;
#pragma unroll
          for (int n = 0; n < 8; ++n)
#pragma unroll
            for (int ey = 0; ey < 3; ++ey)
#pragma unroll
              for (int ex = 0; ex < 3; ++ex)
                s += filtL[ph][n][ey * 3 + ex] * tmpl[n][oh + 2 - ey][ow + 2 - ex];
          float add = ((ph == 0) ? c0 : c1) * s;
          if (pp < 256) racc0 += add; else racc1 += add;
        }
      }
      __syncthreads();
    }

    // accumulate into global (each (t,y,x) owned by exactly one thread here)
    {
      int oh = tid / OW, ow = tid % OW;
      int gy = tileY0 + oh, gx = tileX0 + ow;
      if (gy < HH && gx < WW) {
        size_t idx = ((size_t)t * HH + gy) * WW + gx;
        acc[idx] += racc0;
      }
      int pp2 = tid + 256;
      if (pp2 < OH * OW) {
        oh = pp2 / OW; ow = pp2 % OW;
        gy = tileY0 + oh; gx = tileX0 + ow;
        if (gy < HH && gx < WW) {
          size_t idx = ((size_t)t * HH + gy) * WW + gx;
          acc[idx] += racc1;
        }
      }
    }
  }
}

// ---------------------------- final combine --------------------------------
__global__ __launch_bounds__(256) void vn_final(const float* __restrict__ x0,
                                                const float* __restrict__ acc,
                                                float* __restrict__ out) {
  size_t i = (size_t)blockIdx.x * 256 + threadIdx.x;
  if (i < (size_t)TT * NPIX) out[i] = x0[i] - acc[i];
}

// ---------------------------------------------------------------------------
extern "C" void kernel_launch(void* const* d_in, const int* in_sizes, int n_in,
                              void* d_out, int out_size, void* d_ws, size_t ws_size,
                              hipStream_t stream) {
  (void)in_sizes; (void)n_in; (void)out_size; (void)ws_size;
  const float* x0 = (const float*)d_in[0];
  const float* sC = (const float*)d_in[1];
  const float* mask = (const float*)d_in[2];
  const float* filters = (const float*)d_in[3];
  const float* actv = (const float*)d_in[4];
  const float* alpha = (const float*)d_in[5];
  const float* mu = (const float*)d_in[6];
  float* out = (float*)d_out;

  char* ws = (char*)d_ws;
  float2* cplx = (float2*)ws;                          // 134,217,728 B
  float* phiB = (float*)(ws + 134217728ull);           //  83,886,080 B
  float* acc  = (float*)(ws + 218103808ull);           //  67,108,864 B
  float* coef = (float*)(ws + 285212672ull);           //          64 B

  vn_weights_kernel<<<1, 1, 0, stream>>>(alpha, mu, coef);

  vn_fft_rows_fwd<<<TT * HH, 256, 0, stream>>>(x0, cplx);
  vn_fft_cols_mask<<<TT * WW, 256, 0, stream>>>(cplx, mask, sC);
  vn_ifft_rows_real<<<TT * HH, 256, 0, stream>>>(cplx, coef, acc);

  vn_phi_kernel<<<dim3(WW / 16, HH / 16), 256, 0, stream>>>(x0, filters, actv, phiB);

  dim3 cgrid((WW + OW - 1) / OW, (HH + OH - 1) / OH);
  for (int lp = 0; lp < 5; ++lp)
    vn_conv_pair<<<cgrid, 256, 0, stream>>>(x0, filters, phiB, coef, acc, lp);

  size_t n = (size_t)TT * NPIX;
  vn_final<<<(unsigned)((n + 255) / 256), 256, 0, stream>>>(x0, acc, out);
}